// SimpleModel_20736102105547
// MI455X (gfx1250) — compile-verified
//
#include <hip/hip_runtime.h>
#include <math.h>

typedef float v2f __attribute__((ext_vector_type(2)));
typedef float v8f __attribute__((ext_vector_type(8)));

#define LOG2E 1.4426950408889634f

// Branchless fast transcendentals on the v_exp_f32 / v_rcp_f32 TRANS pipe.
// Endpoint-exact: exp2(+inf)->inf, rcp(inf)->0, exp2(-inf)->0.
__device__ __forceinline__ float fast_rcp(float x) {
#if __has_builtin(__builtin_amdgcn_rcpf)
  return __builtin_amdgcn_rcpf(x);
#else
  return 1.0f / x;
#endif
}
__device__ __forceinline__ float fast_exp2(float x) {
#if __has_builtin(__builtin_amdgcn_exp2f)
  return __builtin_amdgcn_exp2f(x);
#else
  return exp2f(x);
#endif
}
__device__ __forceinline__ float fast_sigmoid(float x) {
  return fast_rcp(1.0f + fast_exp2(-x * LOG2E));
}
__device__ __forceinline__ float fast_tanh(float x) {
#if __has_builtin(__builtin_amdgcn_tanhf)
  return __builtin_amdgcn_tanhf(x);
#else
  // tanh(x) = 1 - 2/(1 + e^{2x})
  return 1.0f - 2.0f * fast_rcp(1.0f + fast_exp2(x * (2.0f * LOG2E)));
#endif
}

// x[t][j] for j in 0..11 (K padded 9 -> 12):
//   j<8 : tanh(W_enc[j]*obs[t] + b_enc[j]),  j==8 : a_prev[t],  j>8 : 0
__device__ __forceinline__ float xelem(int j, float o, float ap,
                                       const float* __restrict__ W_enc,
                                       const float* __restrict__ b_enc) {
  if (j < 8) return fast_tanh(W_enc[j] * o + b_enc[j]);
  if (j == 8) return ap;
  return 0.0f;
}

// W_ih^T element (K row k, output column col), zero-padded past K=9
__device__ __forceinline__ float wihT(const float* __restrict__ W_ih, int k, int col) {
  return (k < 9) ? W_ih[col * 9 + k] : 0.0f;
}

// ---------------------------------------------------------------------------
// Kernel A: GX[t][0..47] = W_ih @ x[t] + b_ih for all t, via f32 WMMA.
// One wave handles 16 timesteps (M=16). N=48 as three 16-wide tiles,
// K=12 as three k-chunks of 4. 9 v_wmma_f32_16x16x4_f32 per wave.
// ---------------------------------------------------------------------------
__global__ __launch_bounds__(128) void gru_gx_wmma(
    const float* __restrict__ obs, const float* __restrict__ acts,
    const float* __restrict__ W_enc, const float* __restrict__ b_enc,
    const float* __restrict__ W_ih, const float* __restrict__ b_ih,
    float* __restrict__ GX, int T, int ntiles) {
  int wave = threadIdx.x >> 5;
  int lane = threadIdx.x & 31;
  int tile = blockIdx.x * 4 + wave;
  if (tile >= ntiles) return;
  int t0 = tile * 16;
  int q = lane & 15;
  bool hi = lane >= 16;

  int t = t0 + q;
  bool tin = t < T;
  float o  = tin ? obs[t] : 0.0f;
  float ap = (tin && t > 0) ? acts[t - 1] : 0.0f;

  // Pre-load bias into the accumulators (same bias for every row M).
  float bi0 = b_ih[q], bi1 = b_ih[16 + q], bi2 = b_ih[32 + q];
  v8f acc0, acc1, acc2;
#pragma unroll
  for (int v = 0; v < 8; ++v) { acc0[v] = bi0; acc1[v] = bi1; acc2[v] = bi2; }

#pragma unroll
  for (int c = 0; c < 3; ++c) {
    int k0 = 4 * c + (hi ? 2 : 0);
    v2f A;
    A.x = xelem(k0,     o, ap, W_enc, b_enc);
    A.y = xelem(k0 + 1, o, ap, W_enc, b_enc);
    v2f B0, B1, B2;
    B0.x = wihT(W_ih, k0, q);          B0.y = wihT(W_ih, k0 + 1, q);
    B1.x = wihT(W_ih, k0, 16 + q);     B1.y = wihT(W_ih, k0 + 1, 16 + q);
    B2.x = wihT(W_ih, k0, 32 + q);     B2.y = wihT(W_ih, k0 + 1, 32 + q);
    acc0 = __builtin_amdgcn_wmma_f32_16x16x4_f32(false, A, false, B0, (short)0, acc0, false, false);
    acc1 = __builtin_amdgcn_wmma_f32_16x16x4_f32(false, A, false, B1, (short)0, acc1, false, false);
    acc2 = __builtin_amdgcn_wmma_f32_16x16x4_f32(false, A, false, B2, (short)0, acc2, false, false);
  }

  int mbase = hi ? 8 : 0;
#pragma unroll
  for (int v = 0; v < 8; ++v) {
    int tt = t0 + mbase + v;
    if (tt < T) {
      float* row = GX + (size_t)tt * 48;
      row[q]      = acc0[v];
      row[16 + q] = acc1[v];
      row[32 + q] = acc2[v];
    }
  }
}

// ---------------------------------------------------------------------------
// Kernel B: the serial scan. One wave32. Lane l owns gate rows l (0..31) and
// 32+(l&15) (n-gate). The new state is broadcast with 16 v_readlane's into
// SGPRs (no LDS round-trip, no waits); the W_hh/W_imp FMAs consume the
// broadcast state as SGPR operands.
// ---------------------------------------------------------------------------
__global__ __launch_bounds__(32) void gru_scan(
    const float* __restrict__ GX, const float* __restrict__ obs,
    const float* __restrict__ acts, const float* __restrict__ bounces,
    const float* __restrict__ W_enc, const float* __restrict__ b_enc,
    const float* __restrict__ W_ih, const float* __restrict__ b_ih,
    const float* __restrict__ W_hh, const float* __restrict__ b_hh,
    const float* __restrict__ W_imp, const float* __restrict__ b_imp,
    float* __restrict__ out, int T, int use_gx) {
  int l = threadIdx.x;
  int lq = l & 15;

  // Recurrent weights in VGPRs: lane l holds W_hh rows l and 32+lq, W_imp row lq.
  float whh0[16], whh1[16], wimp[16];
#pragma unroll
  for (int i = 0; i < 16; ++i) {
    whh0[i] = W_hh[l * 16 + i];
    whh1[i] = W_hh[(32 + lq) * 16 + i];
    wimp[i] = W_imp[lq * 17 + i];
  }
  float bhh0 = b_hh[l], bhh1 = b_hh[32 + lq];
  float wimp16 = W_imp[lq * 17 + 16], bimp = b_imp[lq];

  // Fallback (no workspace) path weights.
  float wih0[9], wih1[9], bih0 = 0.0f, bih1 = 0.0f, we[8], be[8];
  if (!use_gx) {
#pragma unroll
    for (int j = 0; j < 9; ++j) { wih0[j] = W_ih[l * 9 + j]; wih1[j] = W_ih[(32 + lq) * 9 + j]; }
    bih0 = b_ih[l]; bih1 = b_ih[32 + lq];
#pragma unroll
    for (int j = 0; j < 8; ++j) { we[j] = W_enc[j]; be[j] = b_enc[j]; }
  } else {
#pragma unroll
    for (int j = 0; j < 9; ++j) { wih0[j] = 0.0f; wih1[j] = 0.0f; }
#pragma unroll
    for (int j = 0; j < 8; ++j) { we[j] = 0.0f; be[j] = 0.0f; }
  }

  float bb[16];
#pragma unroll
  for (int i = 0; i < 16; ++i) bb[i] = 0.0f;  // b0 = 0
  float b_lane = 0.0f;

  for (int t = 0; t < T; ++t) {
    float ap = (t > 0) ? acts[t - 1] : 0.0f;
    bool cond = (t > 0) && (bounces[t] == 1.0f);

    float gx_lo, gx_hi;
    if (use_gx) {
      const float* row = GX + (size_t)t * 48;
      gx_lo = row[l];
      gx_hi = row[32 + lq];
      if (t + 16 < T) {
        // hide HBM latency behind the serial chain
        __builtin_prefetch(row + 48 * 16, 0, 0);
      }
    } else {
      float o = obs[t];
      float xv[9];
#pragma unroll
      for (int j = 0; j < 8; ++j) xv[j] = fast_tanh(we[j] * o + be[j]);
      xv[8] = ap;
      gx_lo = bih0; gx_hi = bih1;
#pragma unroll
      for (int j = 0; j < 9; ++j) {
        gx_lo = fmaf(wih0[j], xv[j], gx_lo);
        gx_hi = fmaf(wih1[j], xv[j], gx_hi);
      }
    }

    // gh = W_hh @ b + b_hh : lane-parallel FMAs against broadcast (SGPR) state.
    float gh0 = bhh0, gh1 = bhh1;
#pragma unroll
    for (int i = 0; i < 16; ++i) {
      gh0 = fmaf(whh0[i], bb[i], gh0);
      gh1 = fmaf(whh1[i], bb[i], gh1);
    }

    float pre = gx_lo + gh0;                  // lanes 0-15: r-gate, 16-31: z-gate
    float s = fast_sigmoid(pre);
    float zg = __shfl_down(s, 16, 32);        // z-gate aligned to lanes 0-15
    float n = fast_tanh(gx_hi + s * gh1);     // n = tanh(gx_n + r*(W_hh_n@b + b_hh_n))
    float bn = (1.0f - zg) * n + zg * b_lane;

    // impact branch: delta = tanh(W_imp @ [b, ap] + b_imp)
    float dp = fmaf(wimp16, ap, bimp);
#pragma unroll
    for (int i = 0; i < 16; ++i) dp = fmaf(wimp[i], bb[i], dp);
    float bnext = cond ? (b_lane + fast_tanh(dp)) : bn;

    if (l < 16) {
      out[(size_t)t * 16 + l] = bnext;
      b_lane = bnext;
    }

    // Broadcast new state to all lanes via readlane -> SGPRs (no LDS).
#pragma unroll
    for (int i = 0; i < 16; ++i) {
      bb[i] = __uint_as_float(
          __builtin_amdgcn_readlane(__float_as_uint(bnext), i));
    }
  }
}

extern "C" void kernel_launch(void* const* d_in, const int* in_sizes, int n_in,
                              void* d_out, int out_size, void* d_ws, size_t ws_size,
                              hipStream_t stream) {
  const float* obs     = (const float*)d_in[0];
  const float* acts    = (const float*)d_in[1];
  const float* bounces = (const float*)d_in[2];
  const float* W_enc   = (const float*)d_in[3];
  const float* b_enc   = (const float*)d_in[4];
  const float* W_ih    = (const float*)d_in[5];
  const float* b_ih    = (const float*)d_in[6];
  const float* W_hh    = (const float*)d_in[7];
  const float* b_hh    = (const float*)d_in[8];
  const float* W_imp   = (const float*)d_in[9];
  const float* b_imp   = (const float*)d_in[10];
  float* out = (float*)d_out;

  int T = in_sizes[0];
  size_t need = (size_t)T * 48 * sizeof(float);
  float* GX = (float*)d_ws;
  int use_gx = (ws_size >= need) ? 1 : 0;

  if (use_gx) {
    int ntiles = (T + 15) / 16;
    int blocks = (ntiles + 3) / 4;  // 4 waves (tiles) per 128-thread block
    gru_gx_wmma<<<blocks, 128, 0, stream>>>(obs, acts, W_enc, b_enc, W_ih, b_ih,
                                            GX, T, ntiles);
  }
  gru_scan<<<1, 32, 0, stream>>>(use_gx ? GX : nullptr, obs, acts, bounces,
                                 W_enc, b_enc, W_ih, b_ih, W_hh, b_hh,
                                 W_imp, b_imp, out, T, use_gx);
}